// GatbertSelfAttention_15358803050998
// MI455X (gfx1250) — compile-verified
//
#include <hip/hip_runtime.h>
#include <hip/hip_bf16.h>

// ---------------------------------------------------------------------------
// GatbertSelfAttention for MI455X (gfx1250, wave32)
//   B=4, N=4096, HID=128, HEADS=8, DHEAD=16, A=128, E=524288
// Strategy:
//   - f32 WMMA (v_wmma_f32_16x16x4_f32) for every GEMM; one wave owns a
//     16-row x 128-col output tile (8 x v8f accumulators, 32 K-steps).
//   - Edge GEMMs (Ke, Ve) are fused into their consumers: results stay in
//     WMMA accumulators, never hit memory. edge_feats read exactly twice,
//     with NON-TEMPORAL loads so the 268MB stream does not evict the
//     L2-resident gather targets (Q/K/V/out/logits, ~49MB total, L2=192MB).
//   - Segment softmax via monotone-uint atomicMax + atomicAdd (L2-resident).
// ---------------------------------------------------------------------------

#define HIDc   128
#define HEADSc 8
#define Ac     128
#define Bc     4
#define Nc     4096
#define Ec     524288
#define BNc    (Bc * Nc)          // 16384

typedef __attribute__((ext_vector_type(2))) float v2f;
typedef __attribute__((ext_vector_type(8))) float v8f;

// Monotone float<->uint key for atomic max (key 0 == -inf sentinel).
__device__ __forceinline__ unsigned fkey(float f) {
  unsigned u = __float_as_uint(f);
  return (u & 0x80000000u) ? ~u : (u | 0x80000000u);
}
__device__ __forceinline__ float funkey(unsigned u) {
  u = (u & 0x80000000u) ? (u & 0x7fffffffu) : ~u;
  return __uint_as_float(u);
}

// One wave computes acc[16 x 128] += X[row0:row0+16, 0:128] @ W[128 x 128].
// A-operand (16x4 f32): lanes 0-15 hold rows, K={0,1}; lanes 16-31 K={2,3}.
// B-operand (4x16 f32): symmetric layout (N=lane&15, K pair from lane>=16).
// NT=true streams X with non-temporal loads (use for the big edge_feats
// stream); W fragments stay regular-temporal (64KB, L0/L2 hot).
template <bool NT>
__device__ __forceinline__ void wmma_tile_gemm(const float* __restrict__ X, int row0,
                                               const float* __restrict__ W,
                                               v8f acc[8]) {
  const int lane = threadIdx.x & 31;
  const int ar   = lane & 15;           // row within tile (A) / col within tile (B)
  const int koff = (lane >> 4) * 2;     // which K-pair this half-wave supplies
  const float* __restrict__ xrow = X + (size_t)(row0 + ar) * HIDc;
  for (int k = 0; k < HIDc / 4; ++k) {
    v2f a;
    if (NT) a = __builtin_nontemporal_load((const v2f*)(xrow + k * 4 + koff));
    else    a = *(const v2f*)(xrow + k * 4 + koff);
#pragma unroll
    for (int t = 0; t < 8; ++t) {
      v2f b;
      b.x = W[(size_t)(k * 4 + koff    ) * Ac + t * 16 + ar];
      b.y = W[(size_t)(k * 4 + koff + 1) * Ac + t * 16 + ar];
      acc[t] = __builtin_amdgcn_wmma_f32_16x16x4_f32(
          /*neg_a=*/false, a, /*neg_b=*/false, b,
          /*c_mod=*/(short)0, acc[t], /*reuse_a=*/false, /*reuse_b=*/false);
    }
  }
}

// ---------------------------------------------------------------------------
__global__ __launch_bounds__(256) void gat_init_kernel(float* __restrict__ out,
                                                       unsigned* __restrict__ mxkey,
                                                       float* __restrict__ denom) {
  int idx = blockIdx.x * blockDim.x + threadIdx.x;
  if (idx < BNc * Ac) out[idx] = 0.0f;
  if (idx < BNc * HEADSc) { mxkey[idx] = 0u; denom[idx] = 0.0f; }
}

// Y[M x 128] = X[M x 128] @ W[128 x 128] + bias   (one wave per 16 rows)
__global__ __launch_bounds__(128) void gat_proj_kernel(const float* __restrict__ X,
                                                       const float* __restrict__ W,
                                                       const float* __restrict__ bias,
                                                       float* __restrict__ Y, int M) {
  int wave = (blockIdx.x * blockDim.x + threadIdx.x) >> 5;
  int row0 = wave * 16;
  if (row0 >= M) return;                       // wave-uniform
  v8f acc[8] = {};
  wmma_tile_gemm<false>(X, row0, W, acc);
  int lane  = threadIdx.x & 31;
  int n     = lane & 15;
  int mbase = (lane >> 4) * 8;                  // C/D layout: lanes>=16 hold M=8..15
#pragma unroll
  for (int t = 0; t < 8; ++t) {
    float bv = bias[t * 16 + n];
#pragma unroll
    for (int r = 0; r < 8; ++r)
      Y[(size_t)(row0 + mbase + r) * Ac + t * 16 + n] = acc[t][r] + bv;
  }
}

// Per 16-edge wave: Ke tile via WMMA (in-register), then per (edge,head)
// logits = 0.25 * sum_d Q[b,i]*(K[b,j]+Ke+bk); atomicMax segment max.
__global__ __launch_bounds__(128) void gat_edge_logits_kernel(
    const float* __restrict__ EF, const float* __restrict__ Wk,
    const float* __restrict__ bk, const int* __restrict__ eidx,
    const float* __restrict__ Qbuf, const float* __restrict__ Kbuf,
    float* __restrict__ logits, unsigned* __restrict__ mxkey) {
  int wave = (blockIdx.x * blockDim.x + threadIdx.x) >> 5;
  int e0 = wave * 16;
  v8f acc[8] = {};
  wmma_tile_gemm<true>(EF, e0, Wk, acc);       // NT stream of edge_feats
  int lane  = threadIdx.x & 31;
  int n     = lane & 15;
  int mbase = (lane >> 4) * 8;
#pragma unroll
  for (int r = 0; r < 8; ++r) {
    int e = e0 + mbase + r;
    int b = eidx[e], i = eidx[Ec + e], j = eidx[2 * Ec + e];
    size_t qrow = (size_t)(b * Nc + i) * Ac;
    size_t krow = (size_t)(b * Nc + j) * Ac;
#pragma unroll
    for (int t = 0; t < 8; ++t) {
      int c = t * 16 + n;
      float partial = Qbuf[qrow + c] * (Kbuf[krow + c] + acc[t][r] + bk[c]);
      partial += __shfl_xor(partial, 1);
      partial += __shfl_xor(partial, 2);
      partial += __shfl_xor(partial, 4);
      partial += __shfl_xor(partial, 8);       // reduced within 16-lane group
      if (n == 0) {
        float logit = partial * 0.25f;         // 1/sqrt(DHEAD=16)
        logits[(size_t)e * HEADSc + t] = logit;
        atomicMax(&mxkey[(b * Nc + i) * HEADSc + t], fkey(logit));
      }
    }
  }
}

// ex = exp(logit - mx[seg]); denom[seg] += ex   (in-place over logits buffer)
__global__ __launch_bounds__(256) void gat_softmax_ex_kernel(
    const int* __restrict__ eidx, const unsigned* __restrict__ mxkey,
    float* __restrict__ logits, float* __restrict__ denom) {
  int idx = blockIdx.x * blockDim.x + threadIdx.x;   // e*HEADS + h
  if (idx >= Ec * HEADSc) return;
  int e = idx >> 3, h = idx & 7;
  int b = eidx[e], i = eidx[Ec + e];
  float mx = funkey(mxkey[(b * Nc + i) * HEADSc + h]);
  float ex = expf(logits[idx] - mx);
  logits[idx] = ex;
  atomicAdd(&denom[(b * Nc + i) * HEADSc + h], ex);
}

// Per 16-edge wave: Ve tile via WMMA (in-register), attn = ex/denom,
// out[b,i] += attn * (Vn[b,j] + Ve + be) via f32 global atomics.
__global__ __launch_bounds__(128) void gat_edge_scatter_kernel(
    const float* __restrict__ EF, const float* __restrict__ We,
    const float* __restrict__ be, const int* __restrict__ eidx,
    const float* __restrict__ Vbuf, const float* __restrict__ ex,
    const float* __restrict__ denom, float* __restrict__ out) {
  int wave = (blockIdx.x * blockDim.x + threadIdx.x) >> 5;
  int e0 = wave * 16;
  v8f acc[8] = {};
  wmma_tile_gemm<true>(EF, e0, We, acc);       // NT stream of edge_feats
  int lane  = threadIdx.x & 31;
  int n     = lane & 15;
  int mbase = (lane >> 4) * 8;
#pragma unroll
  for (int r = 0; r < 8; ++r) {
    int e = e0 + mbase + r;
    int b = eidx[e], i = eidx[Ec + e], j = eidx[2 * Ec + e];
    size_t vrow = (size_t)(b * Nc + j) * Ac;
    size_t orow = (size_t)(b * Nc + i) * Ac;
    int seg = (b * Nc + i) * HEADSc;
#pragma unroll
    for (int t = 0; t < 8; ++t) {
      int c = t * 16 + n;
      float attn = ex[(size_t)e * HEADSc + t] / denom[seg + t];
      float val  = attn * (Vbuf[vrow + c] + acc[t][r] + be[c]);
      atomicAdd(&out[orow + c], val);
    }
  }
}

// ---------------------------------------------------------------------------
extern "C" void kernel_launch(void* const* d_in, const int* in_sizes, int n_in,
                              void* d_out, int out_size, void* d_ws, size_t ws_size,
                              hipStream_t stream) {
  const float* node_states = (const float*)d_in[0];   // (B,N,HID)
  const float* edge_feats  = (const float*)d_in[1];   // (E,HID)
  const int*   edge_index  = (const int*)  d_in[2];   // (3,E)
  const float* Wq = (const float*)d_in[3];  const float* bq = (const float*)d_in[4];
  const float* Wk = (const float*)d_in[5];  const float* bk = (const float*)d_in[6];
  const float* Wv = (const float*)d_in[7];  const float* bv = (const float*)d_in[8];
  const float* We = (const float*)d_in[9];  const float* be = (const float*)d_in[10];
  float* out = (float*)d_out;

  // Workspace layout (floats): Q | K | V | logits/ex | mxkey(u32) | denom
  float* ws = (float*)d_ws;
  float*    Qbuf   = ws;                                   // 16384*128
  float*    Kbuf   = Qbuf + (size_t)BNc * Ac;
  float*    Vbuf   = Kbuf + (size_t)BNc * Ac;
  float*    logits = Vbuf + (size_t)BNc * Ac;              // E*HEADS
  unsigned* mxkey  = (unsigned*)(logits + (size_t)Ec * HEADSc);
  float*    denom  = (float*)(mxkey + (size_t)BNc * HEADSc);

  // 1) zero out / max-keys / denoms
  gat_init_kernel<<<(BNc * Ac + 255) / 256, 256, 0, stream>>>(out, mxkey, denom);

  // 2) node projections: Q, K, Vn (1024 waves each, 4 waves/block)
  const int projBlocks = (BNc / 16) * 32 / 128;            // 256
  gat_proj_kernel<<<projBlocks, 128, 0, stream>>>(node_states, Wq, bq, Qbuf, BNc);
  gat_proj_kernel<<<projBlocks, 128, 0, stream>>>(node_states, Wk, bk, Kbuf, BNc);
  gat_proj_kernel<<<projBlocks, 128, 0, stream>>>(node_states, Wv, bv, Vbuf, BNc);

  // 3) fused Ke-GEMM + logits + segment max (one wave per 16 edges)
  const int edgeBlocks = (Ec / 16) * 32 / 128;             // 8192
  gat_edge_logits_kernel<<<edgeBlocks, 128, 0, stream>>>(
      edge_feats, Wk, bk, edge_index, Qbuf, Kbuf, logits, mxkey);

  // 4) exp + segment-sum denominators
  gat_softmax_ex_kernel<<<(Ec * HEADSc + 255) / 256, 256, 0, stream>>>(
      edge_index, mxkey, logits, denom);

  // 5) fused Ve-GEMM + attention-weighted scatter
  gat_edge_scatter_kernel<<<edgeBlocks, 128, 0, stream>>>(
      edge_feats, We, be, edge_index, Vbuf, logits, denom, out);
}